// ProjectModel_19911468384434
// MI455X (gfx1250) — compile-verified
//
#include <hip/hip_runtime.h>
#include <hip/hip_bf16.h>
#include <math.h>

typedef float v2f __attribute__((ext_vector_type(2)));
typedef float v8f __attribute__((ext_vector_type(8)));

#define NC 128        // detected corners
#define NCT (NC / 16) // corner tiles of 16

// ---------------------------------------------------------------------------
// Kernel 0: init per-corner running mins (uint bits of +inf)
// ---------------------------------------------------------------------------
__global__ void pm_init_mins(unsigned* gmin) {
    int t = blockIdx.x * blockDim.x + threadIdx.x;
    if (t < NC) gmin[t] = 0x7F800000u; // +inf
}

// ---------------------------------------------------------------------------
// Kernel 1: project points, WMMA 16x16x4 distance tiles, running min
// ---------------------------------------------------------------------------
__launch_bounds__(256)
__global__ void pm_project_min(const float* __restrict__ pnt, int np,
                               const float* __restrict__ cor,
                               const float* __restrict__ cam,
                               const float* __restrict__ theta,
                               const float* __restrict__ vang,
                               const int* __restrict__ hptr,
                               const int* __restrict__ wptr,
                               unsigned* __restrict__ gmin) {
    const float D2R = 3.14159265358979323846f / 180.0f;
    const float H = (float)hptr[0];
    const float W = (float)wptr[0];

    // Camera model constants (uniform; cost amortized over all tiles)
    float a = theta[0] * D2R, b = theta[1] * D2R, c = theta[2] * D2R;
    float sa = sinf(a), ca = cosf(a);
    float sb = sinf(b), cb = cosf(b);
    float sc = sinf(c), cc = cosf(c);
    // R = Rz(c) @ Ry(b) @ Rx(a)
    float r00 = cc * cb, r01 = cc * sb * sa - sc * ca, r02 = cc * sb * ca + sc * sa;
    float r10 = sc * cb, r11 = sc * sb * sa + cc * ca, r12 = sc * sb * ca - cc * sa;
    float r20 = -sb,     r21 = cb * sa,                r22 = cb * ca;
    float f  = -H / (2.0f * tanf(vang[0] * 0.5f * D2R));
    float cx = 0.5f * W - 0.5f;
    float cy = 0.5f * H - 0.5f;
    float camx = cam[0], camy = cam[1], camz = cam[2];

    const int  lane = threadIdx.x & 31;
    const int  li   = lane & 15;     // point/corner slot within tile
    const bool hi   = lane >= 16;    // K=2,3 half of the wave

    // A tiles: 16x4 f32 per 16 corners.
    // lanes 0-15:  v0 = A[M][K=0] = cu,      v1 = A[M][K=1] = cv
    // lanes 16-31: v0 = A[M][K=2] = ||c||^2, v1 = A[M][K=3] = 1
    v2f A[NCT];
#pragma unroll
    for (int t = 0; t < NCT; ++t) {
        float cu = cor[(t * 16 + li) * 2 + 0];
        float cv = cor[(t * 16 + li) * 2 + 1];
        A[t].x = hi ? (cu * cu + cv * cv) : cu;
        A[t].y = hi ? 1.0f                : cv;
    }

    v8f acc[NCT];
#pragma unroll
    for (int t = 0; t < NCT; ++t)
#pragma unroll
        for (int r = 0; r < 8; ++r) acc[t][r] = __builtin_inff();

    const v8f czero = {0.f, 0.f, 0.f, 0.f, 0.f, 0.f, 0.f, 0.f};

    const int  ntiles = (np + 15) >> 4;
    // Wave-uniform tile index (scalarized so loop control stays on SALU,
    // EXEC remains all-ones across the WMMA body as the ISA requires).
    const int wslot = __builtin_amdgcn_readfirstlane((int)(threadIdx.x >> 5));
    const int wpb   = (int)(blockDim.x >> 5);
    long wid = (long)blockIdx.x * wpb + wslot;
    long nw  = (long)gridDim.x * wpb;

    for (long tile = wid; tile < ntiles; tile += nw) {
        int idx = (int)(tile << 4) + li;
        if (idx >= np) idx = np - 1; // duplicate last point; min unaffected
        const float* p = pnt + (size_t)idx * 3;

        // CDNA5 prefetch of next tile (global_prefetch_b8)
        if (tile + nw < ntiles)
            __builtin_prefetch(pnt + ((size_t)((tile + nw) << 4) + li) * 3, 0, 1);

        float px = p[0] - camx, py = p[1] - camy, pz = p[2] - camz;
        float x = r00 * px + r01 * py + r02 * pz;
        float y = r10 * px + r11 * py + r12 * pz;
        float z = r20 * px + r21 * py + r22 * pz;
        float iz = 1.0f / z;
        float u = f * (x * iz) + cx;
        float v = H - (f * (y * iz) + cy); // y-invert

        // B tile: 4x16 f32, N striped across lanes.
        // lanes 0-15:  v0 = B[K=0][N] = -2u, v1 = B[K=1][N] = -2v
        // lanes 16-31: v0 = B[K=2][N] = 1,   v1 = B[K=3][N] = ||p||^2
        v2f B;
        B.x = hi ? 1.0f            : -2.0f * u;
        B.y = hi ? (u * u + v * v) : -2.0f * v;

#pragma unroll
        for (int t = 0; t < NCT; ++t) {
            // D[m][n] = exact squared distance corner m <-> point n
            v8f d = __builtin_amdgcn_wmma_f32_16x16x4_f32(
                false, A[t], false, B, (short)0, czero, false, false);
#pragma unroll
            for (int r = 0; r < 8; ++r) {
                // Bare v_min_num_f32: WMMA output is already canonical (RNE,
                // quiet-NaN propagating per ISA 7.12), so skip the
                // v_max_num_f32 x,x,x canonicalization fminf would force.
                float ar = acc[t][r];
                asm("v_min_num_f32 %0, %1, %2"
                    : "=v"(ar)
                    : "v"(ar), "v"(d[r]));
                acc[t][r] = ar;
            }
        }
    }

    // ---- reduce min over N (lanes within each 16-lane half) -> per corner
    __shared__ unsigned smin[NC];
    for (int i = threadIdx.x; i < NC; i += blockDim.x) smin[i] = 0x7F800000u;
    __syncthreads();

#pragma unroll
    for (int t = 0; t < NCT; ++t) {
#pragma unroll
        for (int r = 0; r < 8; ++r) {
            float v = acc[t][r];
            v = fminf(v, __shfl_xor(v, 1, 32));
            v = fminf(v, __shfl_xor(v, 2, 32));
            v = fminf(v, __shfl_xor(v, 4, 32));
            v = fminf(v, __shfl_xor(v, 8, 32));
            if (li == 0) {
                // C/D layout: VGPR r holds M=r (lanes 0-15) and M=r+8 (16-31)
                int cidx = t * 16 + r + (hi ? 8 : 0);
                // d2 >= 0 (clamp rounding noise) -> uint bit-order == float order
                atomicMin(&smin[cidx], __float_as_uint(fmaxf(v, 0.0f)));
            }
        }
    }
    __syncthreads();

    for (int i = threadIdx.x; i < NC; i += blockDim.x)
        atomicMin(&gmin[i], smin[i]);
}

// ---------------------------------------------------------------------------
// Kernel 2: sum the 128 per-corner mins -> scalar
// ---------------------------------------------------------------------------
__global__ void pm_finalize(const unsigned* __restrict__ gmin,
                            float* __restrict__ out) {
    __shared__ float s[NC];
    int t = threadIdx.x;
    s[t] = __uint_as_float(gmin[t]);
    __syncthreads();
    for (int off = NC / 2; off > 0; off >>= 1) {
        if (t < off) s[t] += s[t + off];
        __syncthreads();
    }
    if (t == 0) out[0] = s[0];
}

// ---------------------------------------------------------------------------
extern "C" void kernel_launch(void* const* d_in, const int* in_sizes, int n_in,
                              void* d_out, int out_size, void* d_ws, size_t ws_size,
                              hipStream_t stream) {
    const float* pnt   = (const float*)d_in[0]; // (N_p, 3)
    const float* cor   = (const float*)d_in[1]; // (128, 2)
    const float* cam   = (const float*)d_in[2]; // (3,)
    const float* theta = (const float*)d_in[3]; // (3,)
    const float* vang  = (const float*)d_in[4]; // (1,)
    const int*   hptr  = (const int*)d_in[5];   // scalar height
    const int*   wptr  = (const int*)d_in[6];   // scalar width

    const int np = in_sizes[0] / 3;
    unsigned* gmin = (unsigned*)d_ws;

    pm_init_mins<<<1, NC, 0, stream>>>(gmin);

    // 480 blocks x 8 waves = 3840 waves; 25000 point-tiles -> ~6.5 tiles/wave,
    // enough latency hiding while staying a single pass over HBM.
    pm_project_min<<<480, 256, 0, stream>>>(pnt, np, cor, cam, theta, vang,
                                            hptr, wptr, gmin);

    pm_finalize<<<1, NC, 0, stream>>>(gmin, (float*)d_out);
}